// myQnn_6382321402203
// MI455X (gfx1250) — compile-verified
//
#include <hip/hip_runtime.h>
#include <math.h>

typedef float v2f __attribute__((ext_vector_type(2)));
typedef float v8f __attribute__((ext_vector_type(8)));

#define B_     8
#define L_     20000
#define EMB_   100
#define HID_   128
#define N3_    384      // 3*HID
#define CHUNK_ 512
#define NCH_   40       // ceil(L/CHUNK)
#define XPITCH 104      // padded pitch for 100-wide embedding rows (keeps b64 loads 8B-aligned)

__device__ __forceinline__ float sigmoidf_(float x) { return 1.0f / (1.0f + __expf(-x)); }

// One block = one (batch, chunk). Computes u = x@W with f32 WMMA (16x16x4, 25 K-steps),
// then a 128-channel parallel / 512-step sequential local scan.
// PASS2==0: record chunk aggregates (F = prod f, C = chunk-local scan end).
// PASS2==1: start from carry, accumulate sum_l r*tanh(c) into pooled sums (atomic f32).
template <int PASS2>
__global__ __launch_bounds__(256) void sru_pass_kernel(
    const int* __restrict__ index, const float* __restrict__ emb,
    const float* __restrict__ Wg, const float* __restrict__ bias,
    float* __restrict__ Fagg, float* __restrict__ Cagg,
    const float* __restrict__ carry, float* __restrict__ pool)
{
    __shared__ float xbuf[16 * XPITCH];   // 16 gathered embedding rows
    __shared__ float ubuf[16 * N3_];      // u tile: 16 positions x 384

    const int tid  = threadIdx.x;
    const int lane = tid & 31;
    const int wave = tid >> 5;
    const int b    = blockIdx.x / NCH_;
    const int ch   = blockIdx.x % NCH_;
    const int base = ch * CHUNK_;

    // pull W (100x384 f32 = 153KB) toward L2/WGP$ once
    if (tid < EMB_) __builtin_prefetch(Wg + tid * N3_, 0, 3);

    float Crun = 0.f, Frun = 1.f, hsum = 0.f, bf = 0.f, br = 0.f;
    if (tid < HID_) {
        bf = bias[tid];
        br = bias[HID_ + tid];
        if (PASS2) Crun = carry[(b * NCH_ + ch) * HID_ + tid];
    }

    const int NT = PASS2 ? 3 : 2;  // N-tiles per wave: pass1 needs cols 0..255, pass2 all 384
    for (int mt = 0; mt < CHUNK_ / 16; ++mt) {
        const int p0 = base + mt * 16;

        // cooperative gather of 16 (sequence-mutated) embedding rows into LDS
        for (int e = tid; e < 16 * EMB_; e += 256) {
            int m = e / EMB_;
            int k = e - m * EMB_;
            int l = p0 + m;
            float v = 0.f;
            if (l < L_) {
                int src = (2 * l <= L_ - 1) ? (L_ - 1 - l) : l;
                int id  = index[(size_t)b * L_ + src];
                v = emb[(size_t)id * EMB_ + k];
            }
            xbuf[m * XPITCH + k] = v;
        }
        __syncthreads();

        // WMMA: A(16x100, LDS) x B(100x16 per N-tile, global W) -> u tile in LDS
        const int m_lane = lane & 15;
        const int khalf  = (lane >> 4) << 1;  // 0 or 2 (K-pair select per ISA A/B layout)
        const int half   = lane >> 4;
        for (int jt = 0; jt < NT; ++jt) {
            const int n0 = (wave * NT + jt) * 16;
            const int n  = n0 + m_lane;
            v8f acc = {};
#pragma unroll
            for (int ks = 0; ks < EMB_ / 4; ++ks) {
                const int kb = 4 * ks + khalf;
                v2f a = *(const v2f*)&xbuf[m_lane * XPITCH + kb];
                v2f bw;
                bw.x = Wg[(size_t)kb * N3_ + n];
                bw.y = Wg[(size_t)(kb + 1) * N3_ + n];
                acc = __builtin_amdgcn_wmma_f32_16x16x4_f32(
                    false, a, false, bw, (short)0, acc, false, false);
            }
#pragma unroll
            for (int v = 0; v < 8; ++v)
                ubuf[(v + 8 * half) * N3_ + n0 + m_lane] = acc[v];
        }
        __syncthreads();

        // sequential recurrence over these 16 positions; 128 channels in parallel (waves 0-3)
        if (tid < HID_) {
            const int k = tid;
            for (int m = 0; m < 16; ++m) {
                const int l = p0 + m;
                if (l < L_) {
                    float xt = ubuf[m * N3_ + k];
                    float fp = ubuf[m * N3_ + HID_ + k];
                    float f  = sigmoidf_(fp + bf);
                    float g  = (1.f - f) * xt;
                    Crun = f * Crun + g;
                    if (PASS2) {
                        float rp = ubuf[m * N3_ + 2 * HID_ + k];
                        float r  = sigmoidf_(rp + br);
                        hsum += r * tanhf(Crun);
                    } else {
                        Frun *= f;
                    }
                }
            }
        }
        __syncthreads();
    }

    if (tid < HID_) {
        if (PASS2) {
            unsafeAtomicAdd(&pool[b * HID_ + tid], hsum);   // global_atomic_add_f32
        } else {
            Fagg[(b * NCH_ + ch) * HID_ + tid] = Frun;
            Cagg[(b * NCH_ + ch) * HID_ + tid] = Crun;
        }
    }
}

// cross-chunk exclusive scan of the linear-recurrence aggregates; also zeroes pooled sums
__global__ void sru_combine_kernel(const float* __restrict__ Fagg,
                                   const float* __restrict__ Cagg,
                                   float* __restrict__ carry,
                                   float* __restrict__ pool)
{
    const int b = blockIdx.x, k = threadIdx.x;
    float c = 0.f;
    for (int i = 0; i < NCH_; ++i) {
        size_t o = (size_t)(b * NCH_ + i) * HID_ + k;
        carry[o] = c;
        c = Fagg[o] * c + Cagg[o];
    }
    pool[b * HID_ + k] = 0.f;
}

// conv(1x1 spatial) + relu + global-max collapse into a few dot products; then the linear head
__global__ __launch_bounds__(512) void head_kernel(
    const float* __restrict__ w1, const float* __restrict__ b1,
    const float* __restrict__ w2, const float* __restrict__ b2,
    const float* __restrict__ w3, const float* __restrict__ b3,
    const float* __restrict__ outW, const float* __restrict__ outb,
    const float* __restrict__ pool, float* __restrict__ out)
{
    __shared__ float pl[B_][2 * HID_];
    __shared__ float feats[B_][48];
    const int tid = threadIdx.x;

    for (int e = tid; e < B_ * 2 * HID_; e += blockDim.x) {
        int b = e / (2 * HID_);
        int c = e % (2 * HID_);
        pl[b][c] = pool[b * HID_ + (c & (HID_ - 1))] * (1.0f / (float)L_);  // hcat=[h,h]; mean over L
    }
    __syncthreads();

    if (tid < B_ * 48) {
        int b = tid / 48, j = tid % 48;
        float best = 0.f;  // relu floor; max(relu(s_i)) == max(0, max(s_i))
        if (j < 16) {          // 5x5 kernel, pad 2, 1x1 in -> only center tap (2,2) survives
            int o = j;
            float s = b1[o];
            for (int i = 0; i < 2 * HID_; ++i) s += pl[b][i] * w1[(o * 2 * HID_ + i) * 25 + 12];
            best = fmaxf(best, s);
        } else if (j < 32) {   // 4x4 kernel, pad 2 -> taps (1..2, 1..2)
            int o = j - 16;
            for (int kh = 1; kh <= 2; ++kh)
                for (int kw = 1; kw <= 2; ++kw) {
                    float s = b2[o];
                    for (int i = 0; i < 2 * HID_; ++i)
                        s += pl[b][i] * w2[(o * 2 * HID_ + i) * 16 + kh * 4 + kw];
                    best = fmaxf(best, s);
                }
        } else {               // 3x3 kernel, pad 2 -> all 9 taps
            int o = j - 32;
            for (int kh = 0; kh < 3; ++kh)
                for (int kw = 0; kw < 3; ++kw) {
                    float s = b3[o];
                    for (int i = 0; i < 2 * HID_; ++i)
                        s += pl[b][i] * w3[(o * 2 * HID_ + i) * 9 + kh * 3 + kw];
                    best = fmaxf(best, s);
                }
        }
        feats[b][j] = best;
    }
    __syncthreads();

    if (tid < B_ * 10) {
        int b = tid / 10, c = tid % 10;
        float s = outb[c];
        for (int j = 0; j < 48; ++j) s += feats[b][j] * outW[j * 10 + c];
        out[b * 10 + c] = s;
    }
}

extern "C" void kernel_launch(void* const* d_in, const int* in_sizes, int n_in,
                              void* d_out, int out_size, void* d_ws, size_t ws_size,
                              hipStream_t stream)
{
    (void)in_sizes; (void)n_in; (void)out_size; (void)ws_size;
    const int*   index = (const int*)  d_in[0];
    const float* emb   = (const float*)d_in[1];
    const float* sruW  = (const float*)d_in[2];
    const float* sruB  = (const float*)d_in[3];
    const float* w1    = (const float*)d_in[4];
    const float* b1    = (const float*)d_in[5];
    const float* w2    = (const float*)d_in[6];
    const float* b2    = (const float*)d_in[7];
    const float* w3    = (const float*)d_in[8];
    const float* b3    = (const float*)d_in[9];
    const float* outW  = (const float*)d_in[10];
    const float* outb  = (const float*)d_in[11];
    float* out = (float*)d_out;

    // workspace layout (~0.5 MB total)
    float* wsF     = (float*)d_ws;
    float* wsC     = wsF     + B_ * NCH_ * HID_;
    float* wsCarry = wsC     + B_ * NCH_ * HID_;
    float* wsPool  = wsCarry + B_ * NCH_ * HID_;

    dim3 grid(B_ * NCH_);
    sru_pass_kernel<0><<<grid, 256, 0, stream>>>(index, emb, sruW, sruB,
                                                 wsF, wsC, wsCarry, wsPool);
    sru_combine_kernel<<<B_, HID_, 0, stream>>>(wsF, wsC, wsCarry, wsPool);
    sru_pass_kernel<1><<<grid, 256, 0, stream>>>(index, emb, sruW, sruB,
                                                 wsF, wsC, wsCarry, wsPool);
    head_kernel<<<1, 512, 0, stream>>>(w1, b1, w2, b2, w3, b3, outW, outb, wsPool, out);
}